// NetGraphSage_9234179686415
// MI455X (gfx1250) — compile-verified
//
#include <hip/hip_runtime.h>
#include <hip/hip_bf16.h>

// ---------------------------------------------------------------------------
// GraphSAGE on MI455X (gfx1250): L2-resident atomic scatter + fp32 WMMA GEMMs
// ---------------------------------------------------------------------------

typedef __attribute__((ext_vector_type(2))) float v2f;
typedef __attribute__((ext_vector_type(8))) float v8f;

__device__ __forceinline__ void atomAddF32(float* p, float v) {
    // native global_atomic_add_f32 (coarse-grained device memory)
    unsafeAtomicAdd(p, v);
}

// ---------------------------------------------------------------------------
// 0) zero accumulators (grid-stride)
// ---------------------------------------------------------------------------
__global__ void zero_kernel(float* __restrict__ p, size_t n) {
    size_t i = (size_t)blockIdx.x * blockDim.x + threadIdx.x;
    size_t stride = (size_t)gridDim.x * blockDim.x;
    for (; i < n; i += stride) p[i] = 0.0f;
}

// ---------------------------------------------------------------------------
// 1) edge scatter, d=128: one wave per edge, lane owns a float4 slice.
//    also accumulates in-degree (lane 0).
// ---------------------------------------------------------------------------
__global__ __launch_bounds__(256) void scatter_d128(
    const float* __restrict__ F, const int* __restrict__ src,
    const int* __restrict__ dst, float* __restrict__ hsum,
    float* __restrict__ deg, int n_edges) {
    int e = blockIdx.x * 8 + (threadIdx.x >> 5);
    if (e >= n_edges) return;
    int lane = threadIdx.x & 31;
    int s = src[e], d = dst[e];
    float4 v = *(const float4*)(F + (size_t)s * 128 + lane * 4);
    float* o = hsum + (size_t)d * 128 + lane * 4;
    atomAddF32(o + 0, v.x);
    atomAddF32(o + 1, v.y);
    atomAddF32(o + 2, v.z);
    atomAddF32(o + 3, v.w);
    if (lane == 0) atomAddF32(deg + d, 1.0f);
}

// ---------------------------------------------------------------------------
// 2) edge scatter, d=64: one wave per edge, lane owns a float2 slice.
// ---------------------------------------------------------------------------
__global__ __launch_bounds__(256) void scatter_d64(
    const float* __restrict__ X, const int* __restrict__ src,
    const int* __restrict__ dst, float* __restrict__ hsum, int n_edges) {
    int e = blockIdx.x * 8 + (threadIdx.x >> 5);
    if (e >= n_edges) return;
    int lane = threadIdx.x & 31;
    int s = src[e], d = dst[e];
    float2 v = *(const float2*)(X + (size_t)s * 64 + lane * 2);
    float* o = hsum + (size_t)d * 64 + lane * 2;
    atomAddF32(o + 0, v.x);
    atomAddF32(o + 1, v.y);
}

// ---------------------------------------------------------------------------
// 3) per-graph node counts
// ---------------------------------------------------------------------------
__global__ void count_nodes(const int* __restrict__ n2g, float* __restrict__ gcnt,
                            int n_nodes) {
    int i = blockIdx.x * blockDim.x + threadIdx.x;
    if (i < n_nodes) atomAddF32(gcnt + n2g[i], 1.0f);
}

// ---------------------------------------------------------------------------
// Branch-free 32-step (K=128) WMMA sweep over one weight matrix.
//   sa   : LDS base for this wave's A rows (column 0 of the K range)
//   W    : weight base, row-major [128..64 x 64]; lane offset pre-applied
// A frag (fp32 16x4): lane l -> row m=l&15, K pair at khalf = 2*(l>>4).
// B frag: lane l -> col, VGPR0 = W[kk][col], VGPR1 = W[kk+1][col].
// ---------------------------------------------------------------------------
__device__ __forceinline__ v8f wmma_sweep_k128(
    const float* __restrict__ sa_base, int lds_stride_elems_unused,
    const float* __restrict__ Wlane, v8f acc, const float* __restrict__ saA) {
    // saA points at &sA[m*stride + khalf]; Wlane points at &W[khalf*64 + col]
    const float* wp = Wlane;
    #pragma unroll 4
    for (int k0 = 0; k0 < 128; k0 += 4) {
        v2f a = *(const v2f*)(saA + k0);
        v2f b;
        b.x = wp[0];
        b.y = wp[64];
        acc = __builtin_amdgcn_wmma_f32_16x16x4_f32(
            false, a, false, b, (short)0, acc, false, false);
        wp += 256;                         // advance 4 K rows of 64 cols
    }
    return acc;
}

// ---------------------------------------------------------------------------
// 4) layer 1: x1 = relu([F | hn] @ [W1s ; W1n]) per 16-node tile.
//    Block = 128 threads = 4 waves; wave nt owns output cols nt*16..nt*16+15.
//    K = 256 as two branch-free K=128 sweeps (self, then neighbor weights).
//    LDS tile 16 x 256 fp32, stride 260 floats (4-bank row skew, 8B aligned).
// ---------------------------------------------------------------------------
__global__ __launch_bounds__(128) void layer1_wmma(
    const float* __restrict__ F, const float* __restrict__ hsum,
    const float* __restrict__ deg, const float* __restrict__ W1s,
    const float* __restrict__ W1n, float* __restrict__ x1, int n_nodes) {
    __shared__ float sA[16 * 260];
    const int tid = threadIdx.x;
    const int tile = blockIdx.x * 16;

    // stage: cols [0,128) = features, cols [128,256) = h_neigh / max(deg,1)
    #pragma unroll
    for (int t = 0; t < 4; ++t) {
        int idx = tid + t * 128;          // 0..511 : 16 rows x 32 float4
        int row = idx >> 5;
        int c4  = (idx & 31) * 4;
        int node = tile + row;
        if (node >= n_nodes) node = n_nodes - 1;
        float4 f = *(const float4*)(F + (size_t)node * 128 + c4);
        *(float4*)(&sA[row * 260 + c4]) = f;
        float dinv = 1.0f / fmaxf(deg[node], 1.0f);
        float4 h = *(const float4*)(hsum + (size_t)node * 128 + c4);
        h.x *= dinv; h.y *= dinv; h.z *= dinv; h.w *= dinv;
        *(float4*)(&sA[row * 260 + 128 + c4]) = h;
    }
    __syncthreads();

    const int lane  = tid & 31;
    const int nt    = tid >> 5;            // N-tile 0..3
    const int m     = lane & 15;           // A row
    const int khalf = (lane >> 4) << 1;    // K sub-offset: 0 or 2
    const int col   = nt * 16 + (lane & 15);

    v8f acc = {};
    // self path: A = sA[:, 0:128), B = W1s
    acc = wmma_sweep_k128(nullptr, 0, W1s + (size_t)khalf * 64 + col, acc,
                          &sA[m * 260 + khalf]);
    // neighbor path: A = sA[:, 128:256), B = W1n
    acc = wmma_sweep_k128(nullptr, 0, W1n + (size_t)khalf * 64 + col, acc,
                          &sA[m * 260 + 128 + khalf]);

    #pragma unroll
    for (int r = 0; r < 8; ++r) {
        int mm = r + ((lane >> 4) ? 8 : 0);   // C/D layout: VGPR r -> row r / r+8
        int node = tile + mm;
        if (node < n_nodes)
            x1[(size_t)node * 64 + col] = fmaxf(acc[r], 0.0f);
    }
}

// ---------------------------------------------------------------------------
// Branch-free K=64 sweep (16 WMMA steps) for layer 2.
// ---------------------------------------------------------------------------
__device__ __forceinline__ v8f wmma_sweep_k64(
    const float* __restrict__ saA, const float* __restrict__ Wlane, v8f acc) {
    const float* wp = Wlane;
    #pragma unroll 4
    for (int k0 = 0; k0 < 64; k0 += 4) {
        v2f a = *(const v2f*)(saA + k0);
        v2f b;
        b.x = wp[0];
        b.y = wp[64];
        acc = __builtin_amdgcn_wmma_f32_16x16x4_f32(
            false, a, false, b, (short)0, acc, false, false);
        wp += 256;
    }
    return acc;
}

// ---------------------------------------------------------------------------
// 5) layer 2: x2 = [x1 | hn2] @ [W2s ; W2n], fused per-graph atomic reduce.
//    K = 128 as two branch-free K=64 sweeps. LDS 16 x 128, stride 132.
// ---------------------------------------------------------------------------
__global__ __launch_bounds__(128) void layer2_wmma(
    const float* __restrict__ x1, const float* __restrict__ hsum2,
    const float* __restrict__ deg, const float* __restrict__ W2s,
    const float* __restrict__ W2n, const int* __restrict__ n2g,
    float* __restrict__ gsum, int n_nodes) {
    __shared__ float sA[16 * 132];
    const int tid = threadIdx.x;
    const int tile = blockIdx.x * 16;

    #pragma unroll
    for (int t = 0; t < 2; ++t) {
        int idx = tid + t * 128;          // 0..255 : 16 rows x 16 float4
        int row = idx >> 4;
        int c4  = (idx & 15) * 4;
        int node = tile + row;
        if (node >= n_nodes) node = n_nodes - 1;
        float4 a = *(const float4*)(x1 + (size_t)node * 64 + c4);
        *(float4*)(&sA[row * 132 + c4]) = a;
        float dinv = 1.0f / fmaxf(deg[node], 1.0f);
        float4 h = *(const float4*)(hsum2 + (size_t)node * 64 + c4);
        h.x *= dinv; h.y *= dinv; h.z *= dinv; h.w *= dinv;
        *(float4*)(&sA[row * 132 + 64 + c4]) = h;
    }
    __syncthreads();

    const int lane  = tid & 31;
    const int nt    = tid >> 5;
    const int m     = lane & 15;
    const int khalf = (lane >> 4) << 1;
    const int col   = nt * 16 + (lane & 15);

    v8f acc = {};
    acc = wmma_sweep_k64(&sA[m * 132 + khalf],      W2s + (size_t)khalf * 64 + col, acc);
    acc = wmma_sweep_k64(&sA[m * 132 + 64 + khalf], W2n + (size_t)khalf * 64 + col, acc);

    #pragma unroll
    for (int r = 0; r < 8; ++r) {
        int mm = r + ((lane >> 4) ? 8 : 0);
        int node = tile + mm;
        if (node < n_nodes) {
            int g = n2g[node];
            atomAddF32(gsum + (size_t)g * 64 + col, acc[r]);
        }
    }
}

// ---------------------------------------------------------------------------
// 6) readout: sigmoid((gsum/cnt) . W_fc)
// ---------------------------------------------------------------------------
__global__ void readout_kernel(const float* __restrict__ gsum,
                               const float* __restrict__ gcnt,
                               const float* __restrict__ Wfc,
                               float* __restrict__ out, int n_graphs) {
    int g = blockIdx.x * blockDim.x + threadIdx.x;
    if (g >= n_graphs) return;
    float inv = 1.0f / fmaxf(gcnt[g], 1.0f);
    float s = 0.0f;
    #pragma unroll
    for (int j = 0; j < 64; ++j) s += gsum[(size_t)g * 64 + j] * Wfc[j];
    s *= inv;
    out[g] = 1.0f / (1.0f + __expf(-s));
}

// ---------------------------------------------------------------------------
extern "C" void kernel_launch(void* const* d_in, const int* in_sizes, int n_in,
                              void* d_out, int out_size, void* d_ws, size_t ws_size,
                              hipStream_t stream) {
    const float* F   = (const float*)d_in[0];
    const float* W1s = (const float*)d_in[1];
    const float* W1n = (const float*)d_in[2];
    const float* W2s = (const float*)d_in[3];
    const float* W2n = (const float*)d_in[4];
    const float* Wfc = (const float*)d_in[5];
    const int*   src = (const int*)d_in[6];
    const int*   dst = (const int*)d_in[7];
    const int*   n2g = (const int*)d_in[8];
    float* out = (float*)d_out;

    const int N = in_sizes[0] / 128;
    const int E = in_sizes[6];
    const int G = out_size;

    // workspace carve-up (fp32): [hsum1 N*128][deg N][hsum2 N*64][gsum G*64][gcnt G][x1 N*64]
    float* hsum1 = (float*)d_ws;
    float* deg   = hsum1 + (size_t)N * 128;
    float* hsum2 = deg + N;
    float* gsum  = hsum2 + (size_t)N * 64;
    float* gcnt  = gsum + (size_t)G * 64;
    float* x1    = gcnt + G;
    size_t zlen  = (size_t)N * 128 + N + (size_t)N * 64 + (size_t)G * 64 + G;

    zero_kernel<<<2048, 256, 0, stream>>>((float*)d_ws, zlen);
    scatter_d128<<<(E + 7) / 8, 256, 0, stream>>>(F, src, dst, hsum1, deg, E);
    count_nodes<<<(N + 255) / 256, 256, 0, stream>>>(n2g, gcnt, N);
    layer1_wmma<<<(N + 15) / 16, 128, 0, stream>>>(F, hsum1, deg, W1s, W1n, x1, N);
    scatter_d64<<<(E + 7) / 8, 256, 0, stream>>>(x1, src, dst, hsum2, E);
    layer2_wmma<<<(N + 15) / 16, 128, 0, stream>>>(x1, hsum2, deg, W2s, W2n, n2g, gsum, N);
    readout_kernel<<<(G + 255) / 256, 256, 0, stream>>>(gsum, gcnt, Wfc, out, G);
}